// OptimalTransportLoss_38654705664019
// MI455X (gfx1250) — compile-verified
//
#include <hip/hip_runtime.h>
#include <cmath>
#include <vector>

typedef float v2f __attribute__((ext_vector_type(2)));
typedef float v8f __attribute__((ext_vector_type(8)));

#define NPTS 4096
#define KDIM 64
#define RPB  64              // rows per block (4 waves x 16 rows)
#define BPT  (NPTS / RPB)    // blocks per task = 64
#define LOGN 8.31776616671934f  // ln(4096)

// ---------------------------------------------------------------------------
// Row L2-normalization: 32 lanes per row, float2 per lane, shfl reduction.
// ---------------------------------------------------------------------------
__global__ __launch_bounds__(256) void ot_normalize_kernel(
    const float* __restrict__ x, const float* __restrict__ y,
    float* __restrict__ xn, float* __restrict__ yn)
{
    const int lane = threadIdx.x & 31;
    const int g    = blockIdx.x * 8 + (threadIdx.x >> 5);
    const float* in;
    float* outp;
    int row;
    if (g < NPTS) { in = x; outp = xn; row = g; }
    else          { in = y; outp = yn; row = g - NPTS; }
    v2f v = *(const v2f*)(in + (size_t)row * KDIM + lane * 2);
    float s = v.x * v.x + v.y * v.y;
    #pragma unroll
    for (int m = 16; m >= 1; m >>= 1) s += __shfl_xor(s, m, 32);
    const float inv = 1.0f / sqrtf(s);
    v2f o; o.x = v.x * inv; o.y = v.y * inv;
    *(v2f*)(outp + (size_t)row * KDIM + lane * 2) = o;
}

__global__ __launch_bounds__(256) void ot_zero_kernel(float* __restrict__ p, int n)
{
    const int i = blockIdx.x * 256 + threadIdx.x;
    if (i < n) p[i] = 0.0f;
}

// ---------------------------------------------------------------------------
// Fused GEMM + streaming softmin (one Sinkhorn half-step for all 4 pairs).
//   out[i] = -eps*logsumexp_j( -ln(N) + pot[j]/eps - C[i][j]/eps )
//   C[i][j] = max(1 - A_i . B_j, 0)      (rows are unit-norm)
// doAvg: 1 -> out = 0.5*(self + softmin)  (symmetric Sinkhorn update)
//        0 -> raw softmin                 (init with zeroed pots / final)
// tasks: 0: f_ba (A=x,B=y,pot=g_ab)  1: g_ab (A=y,B=x,pot=f_ba)
//        2: f_aa (A=x,B=x,pot=f_aa)  3: g_bb (A=y,B=y,pot=g_bb)
// ---------------------------------------------------------------------------
__global__ __launch_bounds__(128) void ot_softmin_kernel(
    const float* __restrict__ xn, const float* __restrict__ yn,
    const float* __restrict__ potCur, float* __restrict__ potNxt,
    float eps, int doAvg)
{
    __shared__ float ldsA[RPB * KDIM];   // 16 KB: this block's A rows

    const int task = blockIdx.x / BPT;
    const int rb   = blockIdx.x % BPT;
    const int tid  = threadIdx.x;
    const int wave = tid >> 5;
    const int lane = tid & 31;
    const int half = lane >> 4;          // which 16-lane half
    const int lr   = lane & 15;

    const float* Arows = (task == 1 || task == 3) ? yn : xn;
    const float* Bcols = (task == 0 || task == 3) ? yn : xn;
    const int pinSel   = (task == 0) ? 1 : (task == 1) ? 0 : task;
    const float* pin   = potCur + pinSel * NPTS;
    const float* self  = potCur + task * NPTS;
    float* outp        = potNxt + task * NPTS;

    // Stage A rows (coalesced float4) into LDS.
    {
        const float4* src = (const float4*)(Arows + (size_t)rb * RPB * KDIM);
        float4* dst = (float4*)ldsA;
        for (int i = tid; i < RPB * KDIM / 4; i += 128) dst[i] = src[i];
    }
    __syncthreads();

    // A operand registers for this wave's 16x64 tile, in WMMA f32 16x16x4
    // layout: lane holds row (lane&15); VGPR pair t covers K = 4t + {2*half, 2*half+1}.
    v2f aK[16];
    {
        const float* arow = ldsA + (wave * 16 + lr) * KDIM + 2 * half;
        #pragma unroll
        for (int t = 0; t < 16; ++t) aK[t] = *(const v2f*)(arow + 4 * t);
    }

    const float invEps = 1.0f / eps;
    float mAcc[8], sAcc[8];
    #pragma unroll
    for (int r = 0; r < 8; ++r) { mAcc[r] = -1e30f; sAcc[r] = 0.0f; }

    const float* bbase = Bcols + (size_t)lr * KDIM + 2 * half;
    // Two column tiles in flight: 2 independent WMMA accumulation chains.
    for (int tile = 0; tile < NPTS / 16; tile += 2) {
        const float* bp0 = bbase + (size_t)tile * (16 * KDIM);
        const float* bp1 = bp0 + 16 * KDIM;
        v8f acc0 = {0.f, 0.f, 0.f, 0.f, 0.f, 0.f, 0.f, 0.f};
        v8f acc1 = {0.f, 0.f, 0.f, 0.f, 0.f, 0.f, 0.f, 0.f};
        #pragma unroll
        for (int t = 0; t < 16; ++t) {
            v2f b0 = *(const v2f*)(bp0 + 4 * t);
            v2f b1 = *(const v2f*)(bp1 + 4 * t);
            acc0 = __builtin_amdgcn_wmma_f32_16x16x4_f32(
                       false, aK[t], false, b0, (short)0, acc0, false, false);
            acc1 = __builtin_amdgcn_wmma_f32_16x16x4_f32(
                       false, aK[t], false, b1, (short)0, acc1, false, false);
        }
        // h[j]/eps for the two columns this lane owns (same for all 8 rows).
        const float h0 = pin[tile * 16 + lr] * invEps;
        const float h1 = pin[tile * 16 + 16 + lr] * invEps;
        // Online logsumexp, single v_exp per element:
        //   d = z - m; e = exp(-|d|);
        //   d<=0:  s += e          (m unchanged)
        //   d> 0:  s = s*e + 1;    m = z
        #pragma unroll
        for (int r = 0; r < 8; ++r) {
            {
                float c = fmaxf(1.0f - acc0[r], 0.0f);
                float z = __fmaf_rn(-c, invEps, h0);
                float d = z - mAcc[r];
                float e = __expf(-fabsf(d));
                bool  g = d > 0.0f;
                sAcc[r] = g ? __fmaf_rn(sAcc[r], e, 1.0f) : (sAcc[r] + e);
                mAcc[r] = g ? z : mAcc[r];
            }
            {
                float c = fmaxf(1.0f - acc1[r], 0.0f);
                float z = __fmaf_rn(-c, invEps, h1);
                float d = z - mAcc[r];
                float e = __expf(-fabsf(d));
                bool  g = d > 0.0f;
                sAcc[r] = g ? __fmaf_rn(sAcc[r], e, 1.0f) : (sAcc[r] + e);
                mAcc[r] = g ? z : mAcc[r];
            }
        }
    }

    // Merge partial LSE states across the 16 lanes of each half-wave.
    #pragma unroll
    for (int msk = 1; msk <= 8; msk <<= 1) {
        #pragma unroll
        for (int r = 0; r < 8; ++r) {
            float mo = __shfl_xor(mAcc[r], msk, 32);
            float so = __shfl_xor(sAcc[r], msk, 32);
            float d  = mo - mAcc[r];
            float e  = __expf(-fabsf(d));
            bool  g  = d > 0.0f;
            sAcc[r]  = g ? __fmaf_rn(sAcc[r], e, so) : __fmaf_rn(so, e, sAcc[r]);
            mAcc[r]  = fmaxf(mAcc[r], mo);
        }
    }

    // C/D layout: VGPR r <-> row r (lanes 0-15) / row 8+r (lanes 16-31).
    const int rowBase = rb * RPB + wave * 16 + half * 8;
    if (lr == 0) {
        #pragma unroll
        for (int r = 0; r < 8; ++r) {
            float val = eps * LOGN - eps * (mAcc[r] + __logf(sAcc[r]));
            if (doAvg) val = 0.5f * (self[rowBase + r] + val);
            outp[rowBase + r] = val;
        }
    }
}

// ---------------------------------------------------------------------------
// Final scalar: mean(f_ba - f_aa) + mean(g_ab - g_bb)
// pot layout: [0]=f_ba [1]=g_ab [2]=f_aa [3]=g_bb, each NPTS floats.
// ---------------------------------------------------------------------------
__global__ __launch_bounds__(256) void ot_reduce_kernel(
    const float* __restrict__ pot, float* __restrict__ out)
{
    __shared__ float red[256];
    float s = 0.0f;
    for (int i = threadIdx.x; i < NPTS; i += 256)
        s += (pot[i] - pot[2 * NPTS + i]) + (pot[NPTS + i] - pot[3 * NPTS + i]);
    red[threadIdx.x] = s;
    __syncthreads();
    for (int w = 128; w >= 1; w >>= 1) {
        if (threadIdx.x < w) red[threadIdx.x] += red[threadIdx.x + w];
        __syncthreads();
    }
    if (threadIdx.x == 0) out[0] = red[0] * (1.0f / NPTS);
}

// ---------------------------------------------------------------------------
extern "C" void kernel_launch(void* const* d_in, const int* in_sizes, int n_in,
                              void* d_out, int out_size, void* d_ws, size_t ws_size,
                              hipStream_t stream)
{
    (void)in_sizes; (void)n_in; (void)out_size; (void)ws_size;
    const float* x = (const float*)d_in[0];
    const float* y = (const float*)d_in[1];
    float* out = (float*)d_out;

    // workspace: xn | yn | potA(4*N) | potB(4*N)  (~2.2 MB total)
    float* ws   = (float*)d_ws;
    float* xn   = ws;
    float* yn   = xn + (size_t)NPTS * KDIM;
    float* potA = yn + (size_t)NPTS * KDIM;
    float* potB = potA + 4 * NPTS;

    // geomloss epsilon schedule: diameter^2 -> blur^2, ratio scaling^2
    std::vector<float> eps;
    {
        const double start = 2.0 * std::log(2.0);
        const double stop  = 2.0 * std::log(0.05);
        const double step  = 2.0 * std::log(0.8);
        eps.push_back(4.0f);
        const int cnt = (int)std::ceil((stop - start) / step);
        for (int i = 0; i < cnt; ++i) eps.push_back((float)std::exp(start + i * step));
        eps.push_back(0.0025f);
    }

    ot_normalize_kernel<<<2 * NPTS / 8, 256, 0, stream>>>(x, y, xn, yn);
    ot_zero_kernel<<<(4 * NPTS + 255) / 256, 256, 0, stream>>>(potA, 4 * NPTS);

    // init: read zeroed potentials from potA, write raw softmin to potB
    ot_softmin_kernel<<<4 * BPT, 128, 0, stream>>>(xn, yn, potA, potB, eps[0], 0);
    float* cur = potB;
    float* nxt = potA;
    // symmetric averaged updates along the eps schedule
    for (size_t i = 0; i < eps.size(); ++i) {
        ot_softmin_kernel<<<4 * BPT, 128, 0, stream>>>(xn, yn, cur, nxt, eps[i], 1);
        float* t = cur; cur = nxt; nxt = t;
    }
    // final extrapolation at blur^2 (no averaging)
    ot_softmin_kernel<<<4 * BPT, 128, 0, stream>>>(xn, yn, cur, nxt, eps.back(), 0);
    { float* t = cur; cur = nxt; nxt = t; }

    ot_reduce_kernel<<<1, 256, 0, stream>>>(cur, out);
}